// create_34789235098112
// MI455X (gfx1250) — compile-verified
//
#include <hip/hip_runtime.h>
#include <stdint.h>

#ifndef __has_builtin
#define __has_builtin(x) 0
#endif

#define OSD_B 64
#define OSD_M 384
#define OSD_N 768
#define OSD_NW 25      // 24 packed data words + 1 augmented word
#define OSD_HPW 24     // packed words per H row (768/32)
#define OSD_NTH 512
#define OSD_SORTN 1024

typedef int osd_v4i __attribute__((ext_vector_type(4)));
typedef __attribute__((address_space(1))) osd_v4i* as1_v4i_p;
typedef __attribute__((address_space(3))) osd_v4i* as3_v4i_p;
typedef __attribute__((address_space(1))) int* as1_i_p;
typedef __attribute__((address_space(3))) int* as3_i_p;

__global__ __launch_bounds__(OSD_NTH)
void osd0_gf2_kernel(const float* __restrict__ llr,
                     const int* __restrict__ synd,
                     const int* __restrict__ H,
                     const int* __restrict__ converge,
                     const int* __restrict__ e_v,
                     float* __restrict__ out)
{
  __shared__ __align__(16) union {
    unsigned long long skey[OSD_SORTN];   // sort keys (8 KB)
    unsigned stage[OSD_SORTN * 2];        // async H staging (8 KB), reused after sort
  } u;
  __shared__ unsigned perm[OSD_N];            // 3 KB
  __shared__ unsigned hpack[OSD_M * OSD_HPW]; // 36 KB bit-packed H
  __shared__ unsigned pivbuf[OSD_NW];
  __shared__ int wavemin[16];
  __shared__ int s_pivot;
  __shared__ unsigned xcol[OSD_N];            // 3 KB solution bits (permuted domain)

  const int b    = (int)blockIdx.x;
  const int tid  = (int)threadIdx.x;
  const int lane = tid & 31;
  const int wid  = tid >> 5;

  // ---- 1: build order-preserving sort keys: (sortable_f32 << 16) | index
  for (int i = tid; i < OSD_SORTN; i += OSD_NTH) {
    unsigned long long key;
    if (i < OSD_N) {
      unsigned ub = __float_as_uint(llr[b * OSD_N + i]);
      unsigned k  = (ub & 0x80000000u) ? ~ub : (ub | 0x80000000u);
      key = (((unsigned long long)k) << 16) | (unsigned)i;
    } else {
      key = ~0ull;  // padding sorts last
    }
    u.skey[i] = key;
  }
  __syncthreads();

  // ---- 2: bitonic sort, 1024 elems, 512 threads
  for (int kk = 2; kk <= OSD_SORTN; kk <<= 1) {
    for (int jj = kk >> 1; jj >= 1; jj >>= 1) {
      int i = ((tid & ~(jj - 1)) << 1) | (tid & (jj - 1));
      int l = i | jj;
      unsigned long long a = u.skey[i];
      unsigned long long c = u.skey[l];
      bool asc = ((i & kk) == 0);
      if ((a > c) == asc) { u.skey[i] = c; u.skey[l] = a; }
      __syncthreads();
    }
  }
  for (int i = tid; i < OSD_N; i += OSD_NTH)
    perm[i] = (unsigned)(u.skey[i] & 0xFFFFull);
  __syncthreads();

  // ---- 3: bit-pack H into LDS. Stage 8KB chunks via CDNA5 async global->LDS,
  //         then pack 32 ints -> 1 word with a wave32 ballot.
  const int chunks = (OSD_M * OSD_N) / 2048;  // 144
  for (int c = 0; c < chunks; ++c) {
#if __has_builtin(__builtin_amdgcn_global_load_async_to_lds_b128)
    __builtin_amdgcn_global_load_async_to_lds_b128(
        (as1_v4i_p)(H + c * 2048 + tid * 4),
        (as3_v4i_p)(&u.stage[tid * 4]), 0, 0);
#if __has_builtin(__builtin_amdgcn_s_wait_asynccnt)
    __builtin_amdgcn_s_wait_asynccnt(0);
#else
    asm volatile("s_wait_asynccnt 0" ::: "memory");
#endif
#elif __has_builtin(__builtin_amdgcn_global_load_async_to_lds_b32)
    for (int t = 0; t < 4; ++t) {
      __builtin_amdgcn_global_load_async_to_lds_b32(
          (as1_i_p)(H + c * 2048 + tid * 4 + t),
          (as3_i_p)(&u.stage[tid * 4 + t]), 0, 0);
    }
#if __has_builtin(__builtin_amdgcn_s_wait_asynccnt)
    __builtin_amdgcn_s_wait_asynccnt(0);
#else
    asm volatile("s_wait_asynccnt 0" ::: "memory");
#endif
#else
    {
      const int4* src = (const int4*)(H + c * 2048);
      ((int4*)u.stage)[tid] = src[tid];
    }
#endif
    __syncthreads();
#pragma unroll
    for (int p = 0; p < 4; ++p) {
      unsigned v = u.stage[p * OSD_NTH + tid];
      unsigned bal = (unsigned)__ballot((v & 1u) != 0);
      if (lane == 0) hpack[c * 64 + p * 16 + wid] = bal;
    }
    __syncthreads();
  }

  // ---- 4: build per-thread augmented row (column-permuted) in registers
  unsigned rw[OSD_NW];
#pragma unroll
  for (int q = 0; q < OSD_NW; ++q) rw[q] = 0u;

  if (tid < OSD_M) {
    for (int w = 0; w < OSD_HPW; ++w) {
      unsigned acc = 0u;
      for (int bit = 0; bit < 32; ++bit) {
        unsigned pj = perm[w * 32 + bit];
        unsigned hw = hpack[tid * OSD_HPW + (int)(pj >> 5)];
        acc |= ((hw >> (pj & 31u)) & 1u) << bit;
      }
      switch (w) {  // constant-index stores keep rw[] in VGPRs (SROA-safe)
        case 0:  rw[0]  = acc; break;  case 1:  rw[1]  = acc; break;
        case 2:  rw[2]  = acc; break;  case 3:  rw[3]  = acc; break;
        case 4:  rw[4]  = acc; break;  case 5:  rw[5]  = acc; break;
        case 6:  rw[6]  = acc; break;  case 7:  rw[7]  = acc; break;
        case 8:  rw[8]  = acc; break;  case 9:  rw[9]  = acc; break;
        case 10: rw[10] = acc; break;  case 11: rw[11] = acc; break;
        case 12: rw[12] = acc; break;  case 13: rw[13] = acc; break;
        case 14: rw[14] = acc; break;  case 15: rw[15] = acc; break;
        case 16: rw[16] = acc; break;  case 17: rw[17] = acc; break;
        case 18: rw[18] = acc; break;  case 19: rw[19] = acc; break;
        case 20: rw[20] = acc; break;  case 21: rw[21] = acc; break;
        case 22: rw[22] = acc; break;  case 23: rw[23] = acc; break;
        default: break;
      }
    }
    rw[24] = (unsigned)(synd[b * OSD_M + tid] & 1);
  }

  for (int i = tid; i < OSD_N; i += OSD_NTH) xcol[i] = 0u;
  __syncthreads();

  // ---- 5: GF(2) Gauss-Jordan. RREF is unique, so lowest-unused-row pivoting
  //         reproduces the reference's swap-based result exactly.
  int used = 0, mypiv = -1;
  unsigned curw = 0u;
  for (int j = 0; j < OSD_N; ++j) {
    const int w = j >> 5;
    if ((j & 31) == 0) {
      switch (w) {
        case 0:  curw = rw[0];  break;  case 1:  curw = rw[1];  break;
        case 2:  curw = rw[2];  break;  case 3:  curw = rw[3];  break;
        case 4:  curw = rw[4];  break;  case 5:  curw = rw[5];  break;
        case 6:  curw = rw[6];  break;  case 7:  curw = rw[7];  break;
        case 8:  curw = rw[8];  break;  case 9:  curw = rw[9];  break;
        case 10: curw = rw[10]; break;  case 11: curw = rw[11]; break;
        case 12: curw = rw[12]; break;  case 13: curw = rw[13]; break;
        case 14: curw = rw[14]; break;  case 15: curw = rw[15]; break;
        case 16: curw = rw[16]; break;  case 17: curw = rw[17]; break;
        case 18: curw = rw[18]; break;  case 19: curw = rw[19]; break;
        case 20: curw = rw[20]; break;  case 21: curw = rw[21]; break;
        case 22: curw = rw[22]; break;  case 23: curw = rw[23]; break;
        default: break;
      }
    }
    const unsigned bmask = 1u << (j & 31);
    const bool elig = (tid < OSD_M) && (!used) && ((curw & bmask) != 0u);
    unsigned bal = (unsigned)__ballot(elig);                 // wave32 ballot
    if (lane == 0) wavemin[wid] = bal ? (tid + __builtin_ctz(bal)) : 0x7fffffff;
    __syncthreads();
    if (tid == 0) {
      int mn = 0x7fffffff;
#pragma unroll
      for (int q = 0; q < 16; ++q) mn = (wavemin[q] < mn) ? wavemin[q] : mn;
      s_pivot = mn;
    }
    __syncthreads();
    const int pr = s_pivot;                                  // uniform
    if (pr != 0x7fffffff) {
      if (tid == pr) {
#pragma unroll
        for (int q = 0; q < OSD_NW; ++q) pivbuf[q] = rw[q];
        used = 1; mypiv = j;
      }
      __syncthreads();
      if ((tid < OSD_M) && (tid != pr) && ((curw & bmask) != 0u)) {
#pragma unroll
        for (int q = 0; q < OSD_NW; ++q) rw[q] ^= pivbuf[q];
        curw ^= pivbuf[w];
      }
      // next pivbuf write is fenced by the next iteration's two barriers
    }
  }

  // ---- 6: read solution off pivot rows, scatter through perm, gate on converge
  if (used) xcol[mypiv] = rw[24] & 1u;
  __syncthreads();
  const int cv = converge[b];
  for (int i = tid; i < OSD_N; i += OSD_NTH) {
    const int col = (int)perm[i];
    const float val = (cv != 0)
        ? (float)((unsigned)e_v[b * OSD_N + col] & 0xFFu)
        : (float)xcol[i];
    out[b * OSD_N + col] = val;
  }
}

extern "C" void kernel_launch(void* const* d_in, const int* in_sizes, int n_in,
                              void* d_out, int out_size, void* d_ws, size_t ws_size,
                              hipStream_t stream) {
  (void)in_sizes; (void)n_in; (void)out_size; (void)d_ws; (void)ws_size;
  const float* llr      = (const float*)d_in[0];
  const int*   synd     = (const int*)d_in[1];
  const int*   H        = (const int*)d_in[2];
  const int*   converge = (const int*)d_in[3];
  const int*   e_v      = (const int*)d_in[4];
  float*       out      = (float*)d_out;
  osd0_gf2_kernel<<<dim3(OSD_B), dim3(OSD_NTH), 0, stream>>>(
      llr, synd, H, converge, e_v, out);
}